// BeamVQ_16509854286145
// MI455X (gfx1250) — compile-verified
//
#include <hip/hip_runtime.h>
#include <hip/hip_bf16.h>
#include <math.h>

// ---------------------------------------------------------------------------
// CDNA5 (gfx1250) VQ-VAE forward. All convs / transposed convs / VQ distance
// matmul are implicit-im2col GEMMs on V_WMMA_F32_16X16X4_F32 (native fp32
// matrix pipe, wave32). Weight tiles are staged to LDS with the CDNA5 async
// copy engine (GLOBAL_LOAD_ASYNC_TO_LDS_B128, ASYNCcnt).
// ---------------------------------------------------------------------------

typedef __attribute__((ext_vector_type(2))) float v2f;
typedef __attribute__((ext_vector_type(8))) float v8f;

__device__ __forceinline__ unsigned lds_off_u32(const void* p)
{
    // generic (flat) shared pointer: low 32 bits are the LDS byte offset
    return (unsigned)(unsigned long long)(uintptr_t)p;
}

__device__ __forceinline__ void wait_asynccnt0()
{
#if defined(__has_builtin)
#if __has_builtin(__builtin_amdgcn_s_wait_asynccnt)
    __builtin_amdgcn_s_wait_asynccnt(0);
    return;
#endif
#endif
    asm volatile("s_wait_asynccnt 0x0" ::: "memory");
}

// ---------------------------------------------------------------------------
// Generic WMMA GEMM conv kernel.
// MODE 0: normal conv, implicit im2col gather (stride/pad), NCHW in/out.
// MODE 1: transposed conv (k=4, s=2, p=1) for one output-parity class (po,pw);
//         weights pre-expanded to [Cout][Cin*4]; columns index (n,oh2,ow2).
// Block tile 64(M) x 64(N) x 16(K), 256 threads = 8 wave32.
// Preconditions (guaranteed by all launches): Kg % 16 == 0, Ng % 64 == 0.
// ---------------------------------------------------------------------------
template <int MODE>
__global__ __launch_bounds__(256) void gemm_conv(
    float* __restrict__ out, const float* __restrict__ in,
    const float* __restrict__ wmat, const float* __restrict__ bias,
    int M, int Ng, int Kg,
    int N, int Cin, int Hin, int Win,
    int Cout, int KH, int KW, int stride, int pad,
    int Ho, int Wo, int relu_in, int relu_out, int po, int pw)
{
    __shared__ __align__(16) float As[64][16];
    __shared__ __align__(16) float Bs[16][68];   // +4 pad de-conflicts banks

    const int tid  = threadIdx.x;
    const int lane = tid & 31;
    const int wv   = tid >> 5;
    const int bm   = blockIdx.y * 64;
    const int bn   = blockIdx.x * 64;

    const int rt = (wv & 3) * 16;        // wave's row-tile offset
    const int c0 = (wv >> 2) * 32;       // wave's first col-tile offset
    const int hl = lane >> 4;            // lane half -> K pair / M half
    const int ml = lane & 15;

    const int HoWo = Ho * Wo;
    const int Ho2 = Ho >> 1, Wo2 = Wo >> 1;
    const int HW2 = Ho2 * Wo2;
    const int KHW = KH * KW;

    // per-thread fixed assignment for the async A stage (16 B per lane)
    const int ar = tid >> 2;             // 0..63 row in A tile
    const int aq = (tid & 3) * 4;        // 0,4,8,12 col in A tile
    const int agm = min(bm + ar, M - 1); // clamp: duplicate rows are never
                                         // stored (epilogue checks m < M)
    const unsigned aLds = lds_off_u32(&As[ar][aq]);

    v8f acc0 = {};
    v8f acc1 = {};

    for (int k0 = 0; k0 < Kg; k0 += 16) {
        // ---- A tile: CDNA5 async copy, global -> LDS, 128b per lane ------
        {
            const float* gsrc = wmat + (size_t)agm * Kg + k0 + aq;
            asm volatile("global_load_async_to_lds_b128 %0, %1, off"
                         :: "v"(aLds), "v"((unsigned long long)(uintptr_t)gsrc)
                         : "memory");
        }
        // ---- B tile: branchless implicit im2col gather -------------------
        {
            const int kr = tid >> 4;          // 0..15
            const int cb = (tid & 15) * 4;    // 0..60
            const int gk = k0 + kr;
#pragma unroll
            for (int j = 0; j < 4; ++j) {
                const int gp = bn + cb + j;   // < Ng (Ng % 64 == 0)
                int n, ih, iw, ci;
                if (MODE == 0) {
                    ci = gk / KHW;
                    const int rem = gk - ci * KHW;
                    const int kh  = rem / KW;
                    const int kw  = rem - kh * KW;
                    n = gp / HoWo;
                    const int hw = gp - n * HoWo;
                    const int oh = hw / Wo;
                    const int ow = hw - oh * Wo;
                    ih = oh * stride - pad + kh;
                    iw = ow * stride - pad + kw;
                } else {
                    ci = gk >> 2;
                    const int t4 = gk & 3;
                    const int ti = t4 >> 1, tj = t4 & 1;
                    n = gp / HW2;
                    const int hw  = gp - n * HW2;
                    const int oh2 = hw / Wo2;
                    const int ow2 = hw - oh2 * Wo2;
                    ih = oh2 + (po == 0 ? -ti : (1 - ti));
                    iw = ow2 + (pw == 0 ? -tj : (1 - tj));
                }
                const bool inb = ((unsigned)ih < (unsigned)Hin) &&
                                 ((unsigned)iw < (unsigned)Win);
                const int ihc = min(max(ih, 0), Hin - 1);
                const int iwc = min(max(iw, 0), Win - 1);
                float v = in[(((size_t)n * Cin + ci) * Hin + ihc) * Win + iwc];
                v = inb ? v : 0.f;
                if (relu_in) v = fmaxf(v, 0.f);
                Bs[kr][cb + j] = v;
            }
        }
        wait_asynccnt0();       // A tile landed in LDS (per-wave ASYNCcnt)
        __syncthreads();        // + DScnt for B stores, cross-wave fence

        // ---- 4 x (K=4) WMMA steps per K-tile, two N tiles per wave -------
#pragma unroll
        for (int kk = 0; kk < 4; ++kk) {
            v2f a, b0, b1;
            // A 16x4: lanes 0-15 hold K={0,1}, lanes 16-31 K={2,3}
            a.x = As[rt + ml][kk * 4 + hl * 2 + 0];
            a.y = As[rt + ml][kk * 4 + hl * 2 + 1];
            // B 4x16: lane n holds col n; VGPR0/1 = K rows per lane half
            b0.x = Bs[kk * 4 + hl * 2 + 0][c0 + ml];
            b0.y = Bs[kk * 4 + hl * 2 + 1][c0 + ml];
            b1.x = Bs[kk * 4 + hl * 2 + 0][c0 + 16 + ml];
            b1.y = Bs[kk * 4 + hl * 2 + 1][c0 + 16 + ml];
            acc0 = __builtin_amdgcn_wmma_f32_16x16x4_f32(
                false, a, false, b0, (short)0, acc0, false, false);
            acc1 = __builtin_amdgcn_wmma_f32_16x16x4_f32(
                false, a, false, b1, (short)0, acc1, false, false);
        }
        __syncthreads();
    }

    // ---- epilogue: bias, optional relu, NCHW (or parity-scatter) store ----
#pragma unroll
    for (int r = 0; r < 8; ++r) {
        const int m = bm + rt + hl * 8 + r;
        if (m >= M) continue;
        const float bv = bias ? bias[m] : 0.f;
#pragma unroll
        for (int c = 0; c < 2; ++c) {
            const int p = bn + c0 + c * 16 + ml;   // < Ng by precondition
            float v = (c == 0 ? acc0[r] : acc1[r]) + bv;
            if (relu_out) v = fmaxf(v, 0.f);
            size_t oidx;
            if (MODE == 0) {
                const int n  = p / HoWo;
                const int hw = p - n * HoWo;
                oidx = ((size_t)n * Cout + m) * HoWo + hw;
            } else {
                const int n   = p / HW2;
                const int hw  = p - n * HW2;
                const int oh2 = hw / Wo2;
                const int ow2 = hw - oh2 * Wo2;
                oidx = ((size_t)n * Cout + m) * HoWo
                     + (size_t)(2 * oh2 + po) * Wo + (2 * ow2 + pw);
            }
            out[oidx] = v;
        }
    }
}

// ---------------------------------------------------------------------------
// BatchNorm (training mode): per-channel mean/var over (N,H,W), then apply.
// ---------------------------------------------------------------------------
__global__ __launch_bounds__(256) void bn_stats(
    const float* __restrict__ x, float* __restrict__ stats,
    int C, int HW, int N)
{
    const int c = blockIdx.x;
    const long total = (long)N * HW;
    float s = 0.f, s2 = 0.f;
    for (long i = threadIdx.x; i < total; i += 256) {
        const int n  = (int)(i / HW);
        const int hw = (int)(i - (long)n * HW);
        const float v = x[((size_t)n * C + c) * HW + hw];
        s += v; s2 += v * v;
    }
    __shared__ float ls[256], ls2[256];
    ls[threadIdx.x] = s; ls2[threadIdx.x] = s2;
    __syncthreads();
    for (int o = 128; o > 0; o >>= 1) {
        if (threadIdx.x < o) {
            ls[threadIdx.x]  += ls[threadIdx.x + o];
            ls2[threadIdx.x] += ls2[threadIdx.x + o];
        }
        __syncthreads();
    }
    if (threadIdx.x == 0) {
        const float m = ls[0] / (float)total;
        stats[c]     = m;
        stats[C + c] = ls2[0] / (float)total - m * m;
    }
}

__global__ __launch_bounds__(256) void bn_apply(
    float* __restrict__ out, const float* __restrict__ x,
    const float* __restrict__ stats, const float* __restrict__ g,
    const float* __restrict__ b, int C, int HW, long total, int addTo)
{
    const long i = (long)blockIdx.x * 256 + threadIdx.x;
    if (i >= total) return;
    const int c = (int)((i / HW) % C);
    const float m = stats[c], v = stats[C + c];
    const float y = g[c] * (x[i] - m) * rsqrtf(v + 1e-5f) + b[c];
    if (addTo) out[i] += y; else out[i] = y;
}

// ---------------------------------------------------------------------------
// VQ helpers
// ---------------------------------------------------------------------------
__global__ __launch_bounds__(256) void zero_f(float* p, int n)
{
    const int i = blockIdx.x * 256 + threadIdx.x;
    if (i < n) p[i] = 0.f;
}

__global__ __launch_bounds__(256) void emb_norms_k(
    const float* __restrict__ emb, float* __restrict__ norms, int K, int D)
{
    const int k = blockIdx.x * 256 + threadIdx.x;
    if (k >= K) return;
    float s = 0.f;
    for (int d = 0; d < D; ++d) { const float e = emb[(size_t)k * D + d]; s += e * e; }
    norms[k] = s;
}

// top-3 smallest of s[k] = ||e_k||^2 - 2*dot[k,p] (||f||^2 is rank-invariant)
__global__ __launch_bounds__(256) void vq_select(
    const float* __restrict__ dot, const float* __restrict__ norms,
    int* __restrict__ idx0, int* __restrict__ idx2, int Kc, int HW, int Np)
{
    const int p = blockIdx.x * 256 + threadIdx.x;
    if (p >= Np) return;
    const int n  = p / HW;
    const int hw = p - n * HW;
    const float* dp = dot + (size_t)n * Kc * HW + hw;
    float s0 = 3.4e38f, s1 = 3.4e38f, s2 = 3.4e38f;
    int i0 = 0, i1 = 0, i2 = 0;
    for (int k = 0; k < Kc; ++k) {
        const float s = norms[k] - 2.f * dp[(size_t)k * HW];
        if (s < s0)      { s2 = s1; i2 = i1; s1 = s0; i1 = i0; s0 = s; i0 = k; }
        else if (s < s1) { s2 = s1; i2 = i1; s1 = s;  i1 = k; }
        else if (s < s2) { s2 = s;  i2 = k; }
    }
    idx0[p] = i0;
    idx2[p] = i2;
}

// quantized = emb[idx0] (straight-through), commitment loss sum, 3rd-k counts
__global__ __launch_bounds__(256) void vq_finalize(
    float* __restrict__ Q, const float* __restrict__ Z,
    const float* __restrict__ emb, const int* __restrict__ idx0,
    const int* __restrict__ idx2, float* __restrict__ lossAcc,
    float* __restrict__ counts, int D, int HW, int Np)
{
    const int p = blockIdx.x * 256 + threadIdx.x;
    if (p >= Np) return;
    const int n  = p / HW;
    const int hw = p - n * HW;
    const float* e = emb + (size_t)idx0[p] * D;
    float s = 0.f;
    for (int d = 0; d < D; ++d) {
        const size_t zi = ((size_t)n * D + d) * HW + hw;
        const float q = e[d];
        const float diff = q - Z[zi];
        s += diff * diff;
        Q[zi] = q;
    }
    atomicAdd(lossAcc, s);
    atomicAdd(counts + idx2[p], 1.f);
}

__global__ __launch_bounds__(256) void vq_scalars(
    float* __restrict__ dout, const float* __restrict__ counts,
    const float* __restrict__ lossAcc, int Kc, int Np, int D, long reconElems)
{
    __shared__ float ls[256];
    float s = 0.f;
    for (int k = threadIdx.x; k < Kc; k += 256) {
        const float a = counts[k] / (float)Np;
        s += a * logf(a + 1e-10f);
    }
    ls[threadIdx.x] = s;
    __syncthreads();
    for (int o = 128; o > 0; o >>= 1) {
        if (threadIdx.x < o) ls[threadIdx.x] += ls[threadIdx.x + o];
        __syncthreads();
    }
    if (threadIdx.x == 0) {
        dout[0] = 0.25f * lossAcc[0] / ((float)Np * (float)D);
        dout[1 + reconElems] = expf(-ls[0]);
    }
}

// ---------------------------------------------------------------------------
// Expand torch ConvTranspose2d weights (Cin, Cout, 4, 4), stride 2 / pad 1,
// into 4 parity-class dense matrices we[cls][Cout][Cin*4].
// ---------------------------------------------------------------------------
__global__ __launch_bounds__(256) void expand_wt(
    const float* __restrict__ w, float* __restrict__ we, int Cin, int Cout)
{
    const int i = blockIdx.x * 256 + threadIdx.x;
    const int per = Cout * Cin * 4;
    if (i >= 4 * per) return;
    const int cls = i / per;
    const int rem = i - cls * per;
    const int co  = rem / (Cin * 4);
    const int r2  = rem - co * (Cin * 4);
    const int ci  = r2 >> 2;
    const int t   = r2 & 3;
    const int ti = t >> 1, tj = t & 1;
    const int po = cls >> 1, pw = cls & 1;
    const int kh = (po == 0) ? (1 + 2 * ti) : (2 * ti);
    const int kw = (pw == 0) ? (1 + 2 * tj) : (2 * tj);
    we[i] = w[(((size_t)ci * Cout + co) * 4 + kh) * 4 + kw];
}

// ---------------------------------------------------------------------------
// Host-side orchestration
// ---------------------------------------------------------------------------
static void launch_conv(hipStream_t s, float* out, const float* in,
                        const float* w, const float* bias,
                        int N, int Cin, int Hin, int Win, int Cout,
                        int KH, int KW, int stride, int pad,
                        int relu_in, int relu_out)
{
    const int Ho = (Hin + 2 * pad - KH) / stride + 1;
    const int Wo = (Win + 2 * pad - KW) / stride + 1;
    const int M = Cout, Ng = N * Ho * Wo, Kg = Cin * KH * KW;
    dim3 grid((Ng + 63) / 64, (M + 63) / 64);
    gemm_conv<0><<<grid, 256, 0, s>>>(out, in, w, bias, M, Ng, Kg,
                                      N, Cin, Hin, Win, Cout, KH, KW, stride, pad,
                                      Ho, Wo, relu_in, relu_out, 0, 0);
}

static void launch_convt(hipStream_t s, float* out, const float* in,
                         const float* wexp, const float* bias,
                         int N, int Cin, int Hin, int Win, int Cout,
                         int relu_in, int relu_out)
{
    const int Ho = Hin * 2, Wo = Win * 2;
    const int Kg = Cin * 4;
    const int Ng = N * (Ho / 2) * (Wo / 2);
    for (int cls = 0; cls < 4; ++cls) {
        dim3 grid((Ng + 63) / 64, (Cout + 63) / 64);
        gemm_conv<1><<<grid, 256, 0, s>>>(out, in,
                                          wexp + (size_t)cls * Cout * Kg, bias,
                                          Cout, Ng, Kg, N, Cin, Hin, Win, Cout,
                                          4, 4, 2, 1, Ho, Wo, relu_in, relu_out,
                                          cls >> 1, cls & 1);
    }
}

static void run_res_stack(hipStream_t s, float* X, float* D, float* E, float* stats,
                          const float* wa, const float* g1, const float* b1,
                          const float* wb, const float* g2, const float* b2,
                          int N, int C, int H, int W, int Cmid)
{
    const int HW = H * W;
    for (int i = 0; i < 2; ++i) {
        const float* wai = wa + (size_t)i * Cmid * C * 9;
        const float* wbi = wb + (size_t)i * C * Cmid;
        // h = bn(conv3x3(relu(X)))
        launch_conv(s, D, X, wai, nullptr, N, C, H, W, Cmid, 3, 3, 1, 1, 1, 0);
        bn_stats<<<Cmid, 256, 0, s>>>(D, stats, Cmid, HW, N);
        const long tD = (long)N * Cmid * HW;
        bn_apply<<<(unsigned)((tD + 255) / 256), 256, 0, s>>>(
            D, D, stats, g1 + i * Cmid, b1 + i * Cmid, Cmid, HW, tD, 0);
        // X += bn(conv1x1(relu(h)))
        launch_conv(s, E, D, wbi, nullptr, N, Cmid, H, W, C, 1, 1, 1, 0, 1, 0);
        bn_stats<<<C, 256, 0, s>>>(E, stats, C, HW, N);
        const long tE = (long)N * C * HW;
        bn_apply<<<(unsigned)((tE + 255) / 256), 256, 0, s>>>(
            X, E, stats, g2 + i * C, b2 + i * C, C, HW, tE, 1);
    }
    // trailing relu(X) is fused into the next GEMM's gather (relu_in=1)
}

extern "C" void kernel_launch(void* const* d_in, const int* in_sizes, int n_in,
                              void* d_out, int out_size, void* d_ws, size_t ws_size,
                              hipStream_t stream)
{
    (void)in_sizes; (void)n_in; (void)out_size; (void)ws_size;
    const float* x     = (const float*)d_in[0];
    const float* p_w1  = (const float*)d_in[1];
    const float* p_b1  = (const float*)d_in[2];
    const float* p_w2  = (const float*)d_in[3];
    const float* p_b2  = (const float*)d_in[4];
    const float* p_w3  = (const float*)d_in[5];
    const float* p_b3  = (const float*)d_in[6];
    const float* pr_wa = (const float*)d_in[7];
    const float* pr_g1 = (const float*)d_in[8];
    const float* pr_b1 = (const float*)d_in[9];
    const float* pr_wb = (const float*)d_in[10];
    const float* pr_g2 = (const float*)d_in[11];
    const float* pr_b2 = (const float*)d_in[12];
    const float* pre_w = (const float*)d_in[13];
    const float* pre_b = (const float*)d_in[14];
    const float* emb   = (const float*)d_in[15];
    const float* d_w1  = (const float*)d_in[16];
    const float* d_b1  = (const float*)d_in[17];
    const float* dr_wa = (const float*)d_in[18];
    const float* dr_g1 = (const float*)d_in[19];
    const float* dr_b1 = (const float*)d_in[20];
    const float* dr_wb = (const float*)d_in[21];
    const float* dr_g2 = (const float*)d_in[22];
    const float* dr_b2 = (const float*)d_in[23];
    const float* dt1_w = (const float*)d_in[24];
    const float* dt1_b = (const float*)d_in[25];
    const float* dt2_w = (const float*)d_in[26];
    const float* dt2_b = (const float*)d_in[27];

    float* out = (float*)d_out;
    float* xrec = out + 1;                      // [16,1,256,256]
    const long reconElems = 16L * 256 * 256;

    // ---- workspace carving (aliasing-aware) ------------------------------
    char* base = (char*)d_ws;
    size_t off = 0;
    auto carve = [&](size_t bytes) -> char* {
        char* p = base + off;
        off += (bytes + 255) & ~(size_t)255;
        return p;
    };
    const size_t HW64 = 64 * 64;     // 4096
    float* DOT  = (float*)carve((size_t)16 * 1024 * HW64 * 4);   // 268 MB
    float* bufA = DOT;               // alias: encoder conv1 out, decoder ct1 out
    float* bufB = (float*)carve((size_t)16 * 128 * HW64 * 4);    // 33.5 MB
    float* bufQ = bufB;              // alias: quantized (bufB dead by then)
    float* bufC = (float*)carve((size_t)16 * 128 * HW64 * 4);
    float* bufD = (float*)carve((size_t)16 * 32  * HW64 * 4);
    float* bufE = (float*)carve((size_t)16 * 128 * HW64 * 4);
    float* bufZ = (float*)carve((size_t)16 * 128 * HW64 * 4);
    float* wexp1 = (float*)carve((size_t)4 * 64 * 512 * 4);
    float* wexp2 = (float*)carve((size_t)4 * 1 * 256 * 4);
    float* norms = (float*)carve(1024 * 4);
    int*   idx0  = (int*)carve(65536 * 4);
    int*   idx2  = (int*)carve(65536 * 4);
    float* stats = (float*)carve(2 * 128 * 4);
    float* counts = (float*)carve(1024 * 4);
    float* lossAcc = (float*)carve(256);

    // ---- encoder ---------------------------------------------------------
    // relu(conv 1->64, k4 s2 p1): 256 -> 128
    launch_conv(stream, bufA, x, p_w1, p_b1, 16, 1, 256, 256, 64, 4, 4, 2, 1, 0, 1);
    // relu(conv 64->128, k4 s2 p1): 128 -> 64
    launch_conv(stream, bufB, bufA, p_w2, p_b2, 16, 64, 128, 128, 128, 4, 4, 2, 1, 0, 1);
    // conv 128->128, k3 s1 p1
    launch_conv(stream, bufC, bufB, p_w3, p_b3, 16, 128, 64, 64, 128, 3, 3, 1, 1, 0, 0);
    // residual stack (2 blocks)
    run_res_stack(stream, bufC, bufD, bufE, stats,
                  pr_wa, pr_g1, pr_b1, pr_wb, pr_g2, pr_b2, 16, 128, 64, 64, 32);
    // pre-VQ 1x1 conv (input relu'd: res_stack trailing relu)
    launch_conv(stream, bufZ, bufC, pre_w, pre_b, 16, 128, 64, 64, 128, 1, 1, 1, 0, 1, 0);

    // ---- vector quantization --------------------------------------------
    emb_norms_k<<<4, 256, 0, stream>>>(emb, norms, 1024, 128);
    // dot[k,p] = emb . z  as a 1x1-conv WMMA GEMM (M=1024, K=128)
    launch_conv(stream, DOT, bufZ, emb, nullptr, 16, 128, 64, 64, 1024, 1, 1, 1, 0, 0, 0);
    zero_f<<<4, 256, 0, stream>>>(counts, 1024);
    zero_f<<<1, 256, 0, stream>>>(lossAcc, 1);
    const int Np = 16 * 64 * 64;
    vq_select<<<Np / 256, 256, 0, stream>>>(DOT, norms, idx0, idx2, 1024, 4096, Np);
    vq_finalize<<<Np / 256, 256, 0, stream>>>(bufQ, bufZ, emb, idx0, idx2,
                                              lossAcc, counts, 128, 4096, Np);
    vq_scalars<<<1, 256, 0, stream>>>(out, counts, lossAcc, 1024, Np, 128, reconElems);

    // ---- decoder ---------------------------------------------------------
    launch_conv(stream, bufC, bufQ, d_w1, d_b1, 16, 128, 64, 64, 128, 3, 3, 1, 1, 0, 0);
    run_res_stack(stream, bufC, bufD, bufE, stats,
                  dr_wa, dr_g1, dr_b1, dr_wb, dr_g2, dr_b2, 16, 128, 64, 64, 32);
    // relu(convT 128->64, k4 s2 p1): 64 -> 128  (input relu fused)
    expand_wt<<<(4 * 64 * 128 * 4 + 255) / 256, 256, 0, stream>>>(dt1_w, wexp1, 128, 64);
    launch_convt(stream, bufA, bufC, wexp1, dt1_b, 16, 128, 64, 64, 64, 1, 1);
    // convT 64->1, k4 s2 p1: 128 -> 256, straight into d_out recon region
    expand_wt<<<(4 * 1 * 64 * 4 + 255) / 256, 256, 0, stream>>>(dt2_w, wexp2, 64, 1);
    launch_convt(stream, xrec, bufA, wexp2, dt2_b, 16, 64, 128, 128, 1, 0, 0);
}